// Camera_8057358647825
// MI455X (gfx1250) — compile-verified
//
#include <hip/hip_runtime.h>
#include <stdint.h>

#define IMG_H 224
#define IMG_W 224
#define NPIX (IMG_H * IMG_W)

typedef __attribute__((ext_vector_type(4))) float v4f;
typedef __attribute__((ext_vector_type(2))) float v2f;

// P = K @ inv(Tcw)[:3], stored column-wise for rows (X,Y) packed + row Z scalar
struct ProjP {
  v2f c0, c1, c2, c3;          // (P00,P10), (P01,P11), (P02,P12), (P03,P13)
  float z0, z1, z2, z3;        // row 2 of P
};

__device__ __forceinline__ v2f fma2(v2f a, v2f b, v2f c) {
#if __has_builtin(__builtin_elementwise_fma)
  return __builtin_elementwise_fma(a, b, c);   // -> v_pk_fma_f32 (VOP3P)
#else
  v2f r; r.x = fmaf(a.x, b.x, c.x); r.y = fmaf(a.y, b.y, c.y); return r;
#endif
}

// ---------------------------------------------------------------------------
// init: keys[pix] = ~0 (sentinel: +inf depth), thread 0 computes P from K,Tcw
// ---------------------------------------------------------------------------
__global__ void cam_init_kernel(const float* __restrict__ K,
                                const float* __restrict__ Tcw,
                                float* __restrict__ Pout,
                                unsigned long long* __restrict__ keys) {
  unsigned gid = blockIdx.x * blockDim.x + threadIdx.x;
  if (gid < NPIX) keys[gid] = ~0ull;
  if (gid == 0) {
    float m[16];
#pragma unroll
    for (int i = 0; i < 16; ++i) m[i] = Tcw[i];
    float inv[16];
    inv[0]  =  m[5]*m[10]*m[15] - m[5]*m[11]*m[14] - m[9]*m[6]*m[15] + m[9]*m[7]*m[14] + m[13]*m[6]*m[11] - m[13]*m[7]*m[10];
    inv[4]  = -m[4]*m[10]*m[15] + m[4]*m[11]*m[14] + m[8]*m[6]*m[15] - m[8]*m[7]*m[14] - m[12]*m[6]*m[11] + m[12]*m[7]*m[10];
    inv[8]  =  m[4]*m[9]*m[15]  - m[4]*m[11]*m[13] - m[8]*m[5]*m[15] + m[8]*m[7]*m[13] + m[12]*m[5]*m[11] - m[12]*m[7]*m[9];
    inv[12] = -m[4]*m[9]*m[14]  + m[4]*m[10]*m[13] + m[8]*m[5]*m[14] - m[8]*m[6]*m[13] - m[12]*m[5]*m[10] + m[12]*m[6]*m[9];
    inv[1]  = -m[1]*m[10]*m[15] + m[1]*m[11]*m[14] + m[9]*m[2]*m[15] - m[9]*m[3]*m[14] - m[13]*m[2]*m[11] + m[13]*m[3]*m[10];
    inv[5]  =  m[0]*m[10]*m[15] - m[0]*m[11]*m[14] - m[8]*m[2]*m[15] + m[8]*m[3]*m[14] + m[12]*m[2]*m[11] - m[12]*m[3]*m[10];
    inv[9]  = -m[0]*m[9]*m[15]  + m[0]*m[11]*m[13] + m[8]*m[1]*m[15] - m[8]*m[3]*m[13] - m[12]*m[1]*m[11] + m[12]*m[3]*m[9];
    inv[13] =  m[0]*m[9]*m[14]  - m[0]*m[10]*m[13] - m[8]*m[1]*m[14] + m[8]*m[2]*m[13] + m[12]*m[1]*m[10] - m[12]*m[2]*m[9];
    inv[2]  =  m[1]*m[6]*m[15]  - m[1]*m[7]*m[14]  - m[5]*m[2]*m[15] + m[5]*m[3]*m[14] + m[13]*m[2]*m[7]  - m[13]*m[3]*m[6];
    inv[6]  = -m[0]*m[6]*m[15]  + m[0]*m[7]*m[14]  + m[4]*m[2]*m[15] - m[4]*m[3]*m[14] - m[12]*m[2]*m[7]  + m[12]*m[3]*m[6];
    inv[10] =  m[0]*m[5]*m[15]  - m[0]*m[7]*m[13]  - m[4]*m[1]*m[15] + m[4]*m[3]*m[13] + m[12]*m[1]*m[7]  - m[12]*m[3]*m[5];
    inv[14] = -m[0]*m[5]*m[14]  + m[0]*m[6]*m[13]  + m[4]*m[1]*m[14] - m[4]*m[2]*m[13] - m[12]*m[1]*m[6]  + m[12]*m[2]*m[5];
    inv[3]  = -m[1]*m[6]*m[11]  + m[1]*m[7]*m[10]  + m[5]*m[2]*m[11] - m[5]*m[3]*m[10] - m[9]*m[2]*m[7]   + m[9]*m[3]*m[6];
    inv[7]  =  m[0]*m[6]*m[11]  - m[0]*m[7]*m[10]  - m[4]*m[2]*m[11] + m[4]*m[3]*m[10] + m[8]*m[2]*m[7]   - m[8]*m[3]*m[6];
    inv[11] = -m[0]*m[5]*m[11]  + m[0]*m[7]*m[9]   + m[4]*m[1]*m[11] - m[4]*m[3]*m[9]  - m[8]*m[1]*m[7]   + m[8]*m[3]*m[5];
    inv[15] =  m[0]*m[5]*m[10]  - m[0]*m[6]*m[9]   - m[4]*m[1]*m[10] + m[4]*m[2]*m[9]  + m[8]*m[1]*m[6]   - m[8]*m[2]*m[5];
    float det  = m[0]*inv[0] + m[1]*inv[4] + m[2]*inv[8] + m[3]*inv[12];
    float rdet = 1.0f / det;
#pragma unroll
    for (int r = 0; r < 3; ++r)
#pragma unroll
      for (int c = 0; c < 4; ++c) {
        float s = 0.0f;
#pragma unroll
        for (int k = 0; k < 3; ++k) s += K[r*3 + k] * (inv[k*4 + c] * rdet);
        Pout[r*4 + c] = s;
      }
  }
}

// ---------------------------------------------------------------------------
// scatter: project point, z-key = (float_bits(Z)<<32)|idx, atomic-min per pixel
// ---------------------------------------------------------------------------
__device__ __forceinline__ void cam_splat(float x, float y, float z, unsigned idx,
                                          const ProjP& P,
                                          unsigned long long* __restrict__ keys) {
  // rows X,Y of the projection, packed -> v_pk_fma_f32
  v2f xx = {x, x}, yy = {y, y}, zz = {z, z};
  v2f XY = fma2(P.c0, xx, fma2(P.c1, yy, fma2(P.c2, zz, P.c3)));
  float Z = fmaf(P.z0, x, fmaf(P.z1, y, fmaf(P.z2, z, P.z3)));

  // X/Z and Y/Z share the denominator: one correctly-rounded reciprocal
  // (IEEE 1/Z) + packed Markstein refinement gives the correctly-rounded
  // quotients in 3 VOP3P ops instead of a second full division expansion.
  // Z in {0, NaN, denorm} -> NaN -> compares false -> invalid (same verdict
  // as the reference's where() masking).
  float rz = 1.0f / Z;
  v2f rzz = {rz, rz};
  v2f Zneg = {-Z, -Z};
  v2f q = XY * rzz;                   // v_pk_mul_f32
  v2f r = fma2(Zneg, q, XY);          // exact residual
  v2f dZ = fma2(r, rzz, q);           // correctly-rounded X/Z, Y/Z

  // constant /224, correctly rounded via the same refinement (packed)
  const v2f rc   = {1.0f / 224.0f, 1.0f / 224.0f};
  const v2f c224 = {-224.0f, -224.0f};
  v2f qq = dZ * rc;
  v2f rr = fma2(c224, qq, dZ);
  v2f n  = fma2(rr, rc, qq);          // n.x = Xn, n.y = Yn (bit-exact w/ ref)

  bool valid = (n.x >= 0.0f) && (n.x < 1.0f) &&
               (n.y >= 0.0f) && (n.y < 1.0f) && (Z >= 0.0f);
  if (!valid) return;
  int yi = (int)floorf(n.x * (float)IMG_W);
  yi = min(max(yi, 0), IMG_W - 1);               // foldable to v_med3_i32
  int xi = (int)floorf(n.y * (float)IMG_H);
  xi = min(max(xi, 0), IMG_H - 1);
  int lin = xi * IMG_W + yi;
  // Z >= 0 -> float bits monotone as u32; index in low bits = deterministic tiebreak
  unsigned long long key =
      ((unsigned long long)__float_as_uint(Z) << 32) | (unsigned long long)idx;
  // test-and-test-and-min: a stale read of a monotone min is >= truth, so
  // skipping when key >= cur is always safe; converts most u64 atomics into
  // L2-resident loads (key buffer is 401KB << 192MB L2).
  unsigned long long cur = keys[lin];
  if (key < cur) atomicMin(&keys[lin], key);
}

__global__ void cam_scatter_kernel(const float* __restrict__ pts,
                                   const float* __restrict__ Pm,
                                   unsigned long long* __restrict__ keys,
                                   unsigned N) {
  ProjP P;
  P.c0 = (v2f){Pm[0], Pm[4]};
  P.c1 = (v2f){Pm[1], Pm[5]};
  P.c2 = (v2f){Pm[2], Pm[6]};
  P.c3 = (v2f){Pm[3], Pm[7]};
  P.z0 = Pm[8];  P.z1 = Pm[9];  P.z2 = Pm[10];  P.z3 = Pm[11];

  unsigned groups = N >> 2;                       // 4 points per iteration
  unsigned stride = blockDim.x * gridDim.x;
  for (unsigned g = blockIdx.x * blockDim.x + threadIdx.x; g < groups; g += stride) {
    // 12 floats = 48B per group, 16B-aligned -> three b128 non-temporal loads
    // (NT: don't let 201MB of streamed points evict the 401KB key buffer from L2)
    const v4f* qv = (const v4f*)(pts + 12ull * g);
    v4f a = __builtin_nontemporal_load(qv);
    v4f b = __builtin_nontemporal_load(qv + 1);
    v4f c = __builtin_nontemporal_load(qv + 2);
    unsigned base = g * 4u;
    cam_splat(a.x, a.y, a.z, base + 0u, P, keys);
    cam_splat(a.w, b.x, b.y, base + 1u, P, keys);
    cam_splat(b.z, b.w, c.x, base + 2u, P, keys);
    cam_splat(c.y, c.z, c.w, base + 3u, P, keys);
  }
  // tail (N not multiple of 4) — deterministic, single lane
  if (blockIdx.x == 0 && threadIdx.x == 0) {
    for (unsigned i = groups * 4u; i < N; ++i) {
      const float* qp = pts + 3ull * i;
      cam_splat(qp[0], qp[1], qp[2], i, P, keys);
    }
  }
}

// ---------------------------------------------------------------------------
// finalize: decode winner key -> depth + color/255; zeros for empty pixels
// ---------------------------------------------------------------------------
__global__ void cam_finalize_kernel(const unsigned long long* __restrict__ keys,
                                    const float* __restrict__ colors,
                                    float* __restrict__ image,
                                    float* __restrict__ depth) {
  unsigned pix = blockIdx.x * blockDim.x + threadIdx.x;
  if (pix >= NPIX) return;
  unsigned long long key = keys[pix];
  float r = 0.0f, g = 0.0f, b = 0.0f, d = 0.0f;
  if (key != ~0ull) {
    unsigned idx = (unsigned)key;
    d = __uint_as_float((unsigned)(key >> 32));
    const float* c = colors + 3ull * idx;
    r = c[0] / 255.0f;
    g = c[1] / 255.0f;
    b = c[2] / 255.0f;
  }
  image[3u * pix + 0u] = r;
  image[3u * pix + 1u] = g;
  image[3u * pix + 2u] = b;
  depth[pix] = d;
}

// ---------------------------------------------------------------------------
extern "C" void kernel_launch(void* const* d_in, const int* in_sizes, int n_in,
                              void* d_out, int out_size, void* d_ws, size_t ws_size,
                              hipStream_t stream) {
  const float* points = (const float*)d_in[0];   // (N,3) f32
  const float* colors = (const float*)d_in[1];   // (N,3) f32
  const float* Kmat   = (const float*)d_in[2];   // (3,3) f32
  const float* Tcw    = (const float*)d_in[3];   // (4,4) f32
  unsigned N = (unsigned)(in_sizes[0] / 3);

  // ws layout: [0,48) P matrix | [64, 64 + NPIX*8) u64 z-keys  (~401 KB)
  float* Pm = (float*)d_ws;
  unsigned long long* keys = (unsigned long long*)((char*)d_ws + 64);

  const int t = 256;                              // 8 wave32 waves per block
  cam_init_kernel<<<(NPIX + t - 1) / t, t, 0, stream>>>(Kmat, Tcw, Pm, keys);
  cam_scatter_kernel<<<4096, t, 0, stream>>>(points, Pm, keys, N);

  float* image = (float*)d_out;                   // 224*224*3 floats
  float* depth = image + 3 * NPIX;                // then 224*224 floats
  cam_finalize_kernel<<<(NPIX + t - 1) / t, t, 0, stream>>>(keys, colors, image, depth);
}